// DNMF_63625645523588
// MI455X (gfx1250) — compile-verified
//
#include <hip/hip_runtime.h>
#include <math.h>
#include <stdint.h>

#define N        2048
#define KC       8
#define ALPHA    0.5f
#define BETA     0.1f
#define GAMMA    0.1f
#define EPS_DEN  1e-15f
#define NS_ITERS 18
#define N_OUTER  2      // NUM_OUTER + 1
#define N_INNER  2
#define N_USTEP  20     // 10 * NUM_INNER
#define KSPLIT   16

// GEMM blocking
#define BM 128
#define BN 64
#define BK 32
#define A_STRIDE 35     // 32 + 3 pad  (odd-ish stride -> conflict-free A frag reads)
#define B_STRIDE 72     // 64 + 8 pad  (2*72 % 64 == 16 -> halves hit disjoint banks)

typedef __attribute__((ext_vector_type(2))) float v2f;
typedef __attribute__((ext_vector_type(8))) float v8f;

#if __has_builtin(__builtin_amdgcn_tensor_load_to_lds) && __has_builtin(__builtin_amdgcn_s_wait_tensorcnt)
#define HAVE_TDM 1
typedef unsigned int v4u __attribute__((ext_vector_type(4)));
typedef int          v8i __attribute__((ext_vector_type(8)));
typedef int          v4i __attribute__((ext_vector_type(4)));

// Issue one 2D tile load (rows x cols fp32, row stride in elements) into LDS
// via the Tensor Data Mover.  pad_interval/pad_amount produce the padded LDS
// row stride directly in hardware.  Tracked by TENSORcnt.
__device__ __forceinline__ void tdm_load_tile_2d(const float* gaddr, unsigned lds_off,
                                                 int tile_cols, int tile_rows,
                                                 int row_stride_elems,
                                                 int pad_interval_code, int pad_amount_code) {
    const unsigned long long ga = (unsigned long long)(uintptr_t)gaddr;
    v4u g0;
    g0.x = 1u;                                            // count=1, user descriptor
    g0.y = lds_off;                                       // lds_addr (bytes)
    g0.z = (unsigned)(ga & 0xffffffffu);                  // global_addr[31:0]
    g0.w = (unsigned)((ga >> 32) & 0x01ffffffu)           // global_addr[56:32]
         | 0x80000000u;                                   // type=2 ("image")
    v8i g1;
    g1[0] = (int)((2u << 16)                              // data_size = 4B
                | (1u << 20)                              // pad_enable
                | ((unsigned)pad_interval_code << 22)
                | ((unsigned)pad_amount_code  << 25));    // workgroup_mask=0 (no cluster)
    g1[1] = (int)(0x01000000u << 16);                     // tensor_dim0[15:0]=0 (dim0=16M)
    g1[2] = (int)((0x01000000u >> 16) | (0u << 16));      // dim0 hi | tensor_dim1 lo (dim1=16M)
    g1[3] = (int)((0x01000000u >> 16) | ((unsigned)tile_cols << 16)); // dim1 hi | tile_dim0
    g1[4] = tile_rows;                                    // tile_dim1 | tile_dim2=0
    g1[5] = row_stride_elems;                             // tensor_dim0_stride[31:0]
    g1[6] = 0;                                            // stride hi | tensor_dim1_stride lo
    g1[7] = 0;
    const v4i gz4 = {0, 0, 0, 0};                         // groups 2/3 unused (2D tensor)
    const v8i gz8 = {0, 0, 0, 0, 0, 0, 0, 0};
    __builtin_amdgcn_tensor_load_to_lds(g0, g1, gz4, gz4, gz8, 0);
}
#else
#define HAVE_TDM 0
#endif

// ---------------------------------------------------------------------------
// Full fp32 WMMA GEMM:  C = A @ B   (square n x n, n % 128 == 0)
// 256 threads (8 waves), 128x64 block tile, BK=32, 2x2 16x16 tiles per wave
// (4 WMMAs per 4 fragment loads).  LDS double-buffered; tiles staged by the
// Tensor Data Mover (async, TENSORcnt) overlapped with WMMA compute.
// ---------------------------------------------------------------------------
__global__ __launch_bounds__(256) void wmma_gemm_nn(const float* __restrict__ A,
                                                    const float* __restrict__ B,
                                                    float* __restrict__ C, int n) {
    __shared__ float As[2 * BM * A_STRIDE];
    __shared__ float Bs[2 * BK * B_STRIDE];
    const int t    = threadIdx.x;
    const int wave = t >> 5;
    const int lane = t & 31;
    const int lm   = lane & 15;
    const int lh   = lane >> 4;
    const int bm0  = blockIdx.y * BM;
    const int bn0  = blockIdx.x * BN;
    const int mw   = (wave >> 1) * 32;   // 0,32,64,96
    const int nw   = (wave & 1) * 32;    // 0,32
    v8f acc00 = {}; v8f acc01 = {}; v8f acc10 = {}; v8f acc11 = {};

    auto compute = [&](int buf) {
        const float* Ab = As + buf * (BM * A_STRIDE);
        const float* Bb = Bs + buf * (BK * B_STRIDE);
#pragma unroll
        for (int kk = 0; kk < BK; kk += 4) {
            const int kf = kk + 2 * lh;
            v2f b0; b0.x = Bb[(kf + 0) * B_STRIDE + nw + lm];
                    b0.y = Bb[(kf + 1) * B_STRIDE + nw + lm];
            v2f b1; b1.x = Bb[(kf + 0) * B_STRIDE + nw + 16 + lm];
                    b1.y = Bb[(kf + 1) * B_STRIDE + nw + 16 + lm];
            v2f a0; a0.x = Ab[(mw + lm) * A_STRIDE + kf + 0];
                    a0.y = Ab[(mw + lm) * A_STRIDE + kf + 1];
            v2f a1; a1.x = Ab[(mw + 16 + lm) * A_STRIDE + kf + 0];
                    a1.y = Ab[(mw + 16 + lm) * A_STRIDE + kf + 1];
            acc00 = __builtin_amdgcn_wmma_f32_16x16x4_f32(false, a0, false, b0, (short)0, acc00, false, false);
            acc01 = __builtin_amdgcn_wmma_f32_16x16x4_f32(false, a0, false, b1, (short)0, acc01, false, false);
            acc10 = __builtin_amdgcn_wmma_f32_16x16x4_f32(false, a1, false, b0, (short)0, acc10, false, false);
            acc11 = __builtin_amdgcn_wmma_f32_16x16x4_f32(false, a1, false, b1, (short)0, acc11, false, false);
        }
    };

#if HAVE_TDM
    // --- TDM-staged, LDS double-buffered pipeline ---
    if (t < 32) {   // wave 0 drives the DMA (uniform branch: whole waves skip)
        tdm_load_tile_2d(A + (size_t)bm0 * n, (unsigned)(uintptr_t)&As[0],
                         BK, BM, n, /*interval 32dw*/4, /*pad 3dw*/2);
        tdm_load_tile_2d(B + bn0,             (unsigned)(uintptr_t)&Bs[0],
                         BN, BK, n, /*interval 64dw*/5, /*pad 8dw*/7);
        __builtin_amdgcn_s_wait_tensorcnt(0);
    }
    __syncthreads();
    int buf = 0;
    for (int k0 = 0; k0 < n; k0 += BK) {
        const bool more = (k0 + BK) < n;
        if (more && t < 32) {   // prefetch next tiles while everyone computes
            tdm_load_tile_2d(A + (size_t)bm0 * n + (k0 + BK),
                             (unsigned)(uintptr_t)&As[(buf ^ 1) * (BM * A_STRIDE)],
                             BK, BM, n, 4, 2);
            tdm_load_tile_2d(B + (size_t)(k0 + BK) * n + bn0,
                             (unsigned)(uintptr_t)&Bs[(buf ^ 1) * (BK * B_STRIDE)],
                             BN, BK, n, 5, 7);
        }
        compute(buf);
        if (more && t < 32) __builtin_amdgcn_s_wait_tensorcnt(0);
        __syncthreads();
        buf ^= 1;
    }
#else
    // --- fallback: per-lane copies, register-prefetch + LDS double buffer ---
    float4 ra[4]; float4 rb[2];
    auto load_tiles = [&](int k0) {
#pragma unroll
        for (int q = 0; q < 4; ++q) {
            const int s4 = t + 256 * q;                       // 1024 float4 slots
            const int row = s4 >> 3, col = (s4 & 7) * 4;      // 128 x 32
            ra[q] = *(const float4*)(A + (size_t)(bm0 + row) * n + k0 + col);
        }
#pragma unroll
        for (int q = 0; q < 2; ++q) {
            const int s4 = t + 256 * q;                       // 512 float4 slots
            const int row = s4 >> 4, col = (s4 & 15) * 4;     // 32 x 64
            rb[q] = *(const float4*)(B + (size_t)(k0 + row) * n + bn0 + col);
        }
    };
    auto store_tiles = [&](int buf) {
        float* Ab = As + buf * (BM * A_STRIDE);
        float* Bb = Bs + buf * (BK * B_STRIDE);
#pragma unroll
        for (int q = 0; q < 4; ++q) {
            const int s4 = t + 256 * q;
            const int row = s4 >> 3, col = (s4 & 7) * 4;
            Ab[row * A_STRIDE + col + 0] = ra[q].x; Ab[row * A_STRIDE + col + 1] = ra[q].y;
            Ab[row * A_STRIDE + col + 2] = ra[q].z; Ab[row * A_STRIDE + col + 3] = ra[q].w;
        }
#pragma unroll
        for (int q = 0; q < 2; ++q) {
            const int s4 = t + 256 * q;
            const int row = s4 >> 4, col = (s4 & 15) * 4;
            Bb[row * B_STRIDE + col + 0] = rb[q].x; Bb[row * B_STRIDE + col + 1] = rb[q].y;
            Bb[row * B_STRIDE + col + 2] = rb[q].z; Bb[row * B_STRIDE + col + 3] = rb[q].w;
        }
    };
    load_tiles(0);
    store_tiles(0);
    __syncthreads();
    int buf = 0;
    for (int k0 = 0; k0 < n; k0 += BK) {
        const bool more = (k0 + BK) < n;
        if (more) load_tiles(k0 + BK);
        compute(buf);
        if (more) store_tiles(buf ^ 1);
        __syncthreads();
        buf ^= 1;
    }
#endif

#pragma unroll
    for (int r = 0; r < 8; ++r) {
        const int row = bm0 + mw + r + 8 * lh;
        const int col = bn0 + nw + lm;
        C[(size_t)row * n + col]             = acc00[r];
        C[(size_t)row * n + col + 16]        = acc01[r];
        C[(size_t)(row + 16) * n + col]      = acc10[r];
        C[(size_t)(row + 16) * n + col + 16] = acc11[r];
    }
}

// ---------------------------------------------------------------------------
// Skinny WMMA GEMM with deterministic split-K:
//   Cpart[split] (n x 8) = A[rows, chunk] @ B[chunk, 8]  (B zero-padded to 16)
// ---------------------------------------------------------------------------
__global__ __launch_bounds__(32) void wmma_skinny_gemm(const float* __restrict__ A,
                                                       const float* __restrict__ B,
                                                       float* __restrict__ Cpart, int n) {
    const int lane = threadIdx.x & 31;
    const int lm   = lane & 15;
    const int lh   = lane >> 4;
    const int row0 = blockIdx.x * 16;
    const int chunk = n / KSPLIT;
    const int kbeg  = blockIdx.y * chunk;
    v8f acc = {};
    for (int k = kbeg; k < kbeg + chunk; k += 4) {
        v2f a; a.x = A[(size_t)(row0 + lm) * n + k + 2 * lh];
               a.y = A[(size_t)(row0 + lm) * n + k + 2 * lh + 1];
        v2f b;
        if (lm < KC) { b.x = B[(size_t)(k + 2 * lh) * KC + lm];
                       b.y = B[(size_t)(k + 2 * lh + 1) * KC + lm]; }
        else         { b.x = 0.0f; b.y = 0.0f; }
        acc = __builtin_amdgcn_wmma_f32_16x16x4_f32(false, a, false, b, (short)0, acc, false, false);
    }
    float* Cp = Cpart + (size_t)blockIdx.y * ((size_t)n * KC);
    if (lm < KC) {
#pragma unroll
        for (int r = 0; r < 8; ++r)
            Cp[(size_t)(row0 + r + 8 * lh) * KC + lm] = acc[r];
    }
}

__global__ void skinny_reduce_kernel(const float* __restrict__ Cp,
                                     float* __restrict__ C, int count) {
    const int i = blockIdx.x * 256 + threadIdx.x;
    if (i >= count) return;
    float a = 0.0f;
#pragma unroll
    for (int s = 0; s < KSPLIT; ++s) a += Cp[(size_t)s * count + i];
    C[i] = a;
}

// ---------------------------------------------------------------------------
// Reductions.  mode: 0 = sum of squares, 1 = plain sum, 2 = sum of |x|
// ---------------------------------------------------------------------------
__global__ void row_reduce_kernel(const float* __restrict__ A, float* __restrict__ out,
                                  int n, int mode, float scale) {
    __shared__ float sd[256];
    const int row = blockIdx.x;
    float acc = 0.0f;
    for (int j = threadIdx.x; j < n; j += 256) {
        const float v = A[(size_t)row * n + j];
        acc += (mode == 0) ? v * v : (mode == 2 ? fabsf(v) : v);
    }
    sd[threadIdx.x] = acc; __syncthreads();
    for (int s = 128; s > 0; s >>= 1) {
        if (threadIdx.x < s) sd[threadIdx.x] += sd[threadIdx.x + s];
        __syncthreads();
    }
    if (threadIdx.x == 0) out[row] = sd[0] * scale;
}

__global__ void col_reduce_kernel(const float* __restrict__ A, float* __restrict__ out,
                                  int n, int mode, float scale) {
    const int col = blockIdx.x * 256 + threadIdx.x;
    if (col >= n) return;
    float acc = 0.0f;
    for (int i = 0; i < n; ++i) {
        const float v = A[(size_t)i * n + col];
        acc += (mode == 0) ? v * v : (mode == 2 ? fabsf(v) : v);
    }
    out[col] = acc * scale;
}

// mode: 0 = sum*scale, 1 = max
__global__ void vec_reduce_kernel(const float* __restrict__ v, float* __restrict__ out,
                                  int n, int mode, float scale) {
    __shared__ float sd[256];
    float acc = (mode == 1) ? -3.4e38f : 0.0f;
    for (int i = threadIdx.x; i < n; i += 256) {
        const float x = v[i];
        acc = (mode == 1) ? fmaxf(acc, x) : acc + x;
    }
    sd[threadIdx.x] = acc; __syncthreads();
    for (int s = 128; s > 0; s >>= 1) {
        if (threadIdx.x < s)
            sd[threadIdx.x] = (mode == 1) ? fmaxf(sd[threadIdx.x], sd[threadIdx.x + s])
                                          : sd[threadIdx.x] + sd[threadIdx.x + s];
        __syncthreads();
    }
    if (threadIdx.x == 0) out[0] = sd[0] * scale;
}

__global__ void ns_scale_kernel(const float* __restrict__ scal_in, float* __restrict__ s_out) {
    if (threadIdx.x == 0)
        s_out[0] = 1.0f / fmaxf(scal_in[1] * scal_in[2], 1e-30f);
}

// ---------------------------------------------------------------------------
// Elementwise n x n kernels (block 16x16, grid n/16 x n/16)
// ---------------------------------------------------------------------------
__global__ void kexp_kernel(float* __restrict__ KM, const float* __restrict__ rs,
                            const float* __restrict__ cs, int n) {
    const int j = blockIdx.x * 16 + threadIdx.x;
    const int i = blockIdx.y * 16 + threadIdx.y;
    const size_t idx = (size_t)i * n + j;
    KM[idx] = expf(-0.5f * (rs[i] + cs[j] - 2.0f * KM[idx]));
}

__global__ void center_kernel(float* __restrict__ KM, const float* __restrict__ rm,
                              const float* __restrict__ cm, const float* __restrict__ g, int n) {
    const int j = blockIdx.x * 16 + threadIdx.x;
    const int i = blockIdx.y * 16 + threadIdx.y;
    const size_t idx = (size_t)i * n + j;
    float v = KM[idx] - rm[i] - cm[j] + g[0];
    if (i == j) v += GAMMA;
    KM[idx] = v;
}

__global__ void y_kernel(float* __restrict__ T, int n) {
    const int j = blockIdx.x * 16 + threadIdx.x;
    const int i = blockIdx.y * 16 + threadIdx.y;
    const size_t idx = (size_t)i * n + j;
    float v = -T[idx];
    if (i == j) v += 2.0f;
    T[idx] = v;
}

__global__ void sprime_kernel(const float* __restrict__ Minv, float* __restrict__ SP,
                              float* __restrict__ SPT, float* __restrict__ dsp, int n) {
    const int j = blockIdx.x * 16 + threadIdx.x;
    const int i = blockIdx.y * 16 + threadIdx.y;
    float v = BETA * (GAMMA * Minv[(size_t)i * n + j] - 1.0f / (float)n);
    if (i == j) { v += ALPHA; dsp[i] = v; }
    SP [(size_t)i * n + j] = v;
    SPT[(size_t)j * n + i] = v;
}

__global__ void transpose_scale_kernel(const float* __restrict__ M, const float* __restrict__ s,
                                       float* __restrict__ X, int n) {
    __shared__ float tile[32][33];
    const int x = blockIdx.x * 32 + threadIdx.x;
    const int y = blockIdx.y * 32 + threadIdx.y;
#pragma unroll
    for (int k = 0; k < 32; k += 8)
        tile[threadIdx.y + k][threadIdx.x] = M[(size_t)(y + k) * n + x];
    __syncthreads();
    const float sc = s[0];
    const int xo = blockIdx.y * 32 + threadIdx.x;
    const int yo = blockIdx.x * 32 + threadIdx.y;
#pragma unroll
    for (int k = 0; k < 32; k += 8)
        X[(size_t)(yo + k) * n + xo] = sc * tile[threadIdx.x][threadIdx.y + k];
}

// ---------------------------------------------------------------------------
// Small (K=8) helper kernels
// ---------------------------------------------------------------------------
__global__ void qpm_kernel(const float* __restrict__ Q, float* __restrict__ Qp,
                           float* __restrict__ Qm) {
    const int i = threadIdx.x;  // 64 threads
    const float q = Q[i], a = fabsf(q);
    Qp[i] = 0.5f * (a + q);
    Qm[i] = 0.5f * (a - q);
}

__global__ void fq_kernel(const float* __restrict__ F, const float* __restrict__ Qp,
                          const float* __restrict__ Qm, float* __restrict__ FQp,
                          float* __restrict__ FQm, int n) {
    const int i = blockIdx.x * 256 + threadIdx.x;
    if (i >= n) return;
    float f[KC];
#pragma unroll
    for (int l = 0; l < KC; ++l) f[l] = F[(size_t)i * KC + l];
#pragma unroll
    for (int k = 0; k < KC; ++k) {
        float sp = 0.0f, sm = 0.0f;
#pragma unroll
        for (int l = 0; l < KC; ++l) { sp += f[l] * Qp[l * KC + k]; sm += f[l] * Qm[l * KC + k]; }
        FQp[(size_t)i * KC + k] = sp;
        FQm[(size_t)i * KC + k] = sm;
    }
}

__global__ void atb8_kernel(const float* __restrict__ X, const float* __restrict__ Y,
                            float* __restrict__ out, int n) {
    const int a = threadIdx.x & 7, b = threadIdx.x >> 3;
    float acc = 0.0f;
    for (int i = 0; i < n; ++i) acc += X[(size_t)i * KC + a] * Y[(size_t)i * KC + b];
    out[a * KC + b] = acc;
}

__global__ void u_update_kernel(const float* __restrict__ Uc, const float* __restrict__ AU,
                                const float* __restrict__ FQp, const float* __restrict__ FQm,
                                const float* __restrict__ UtU, float* __restrict__ Un, int n) {
    const int t = blockIdx.x * 256 + threadIdx.x;
    if (t >= n * KC) return;
    const int i = t >> 3, k = t & 7;
    float uu = 0.0f;
#pragma unroll
    for (int l = 0; l < KC; ++l) uu += Uc[(size_t)i * KC + l] * UtU[l * KC + k];
    const float u   = Uc[t];
    const float num = 2.0f * AU[t] + ALPHA * FQp[t];
    float den = 2.0f * uu + ALPHA * u + ALPHA * FQm[t];
    den = fmaxf(den, EPS_DEN);
    const float r = num / den;
    Un[t] = u * sqrtf(sqrtf(r));
}

__global__ void uq_kernel(const float* __restrict__ U, const float* __restrict__ Q,
                          float* __restrict__ UQ, int n) {
    const int t = blockIdx.x * 256 + threadIdx.x;
    if (t >= n * KC) return;
    const int i = t >> 3, k = t & 7;
    float acc = 0.0f;
#pragma unroll
    for (int l = 0; l < KC; ++l) acc += U[(size_t)i * KC + l] * Q[k * KC + l];
    UQ[t] = acc;
}

__global__ void polar8_kernel(const float* __restrict__ Min, float* __restrict__ Qout,
                              int transpose_out) {
    if (threadIdx.x != 0 || blockIdx.x != 0) return;
    float X[64];
    for (int i = 0; i < 64; ++i) X[i] = Min[i];
    for (int it = 0; it < 30; ++it) {
        float W[64], Inv[64];
        for (int i = 0; i < 64; ++i) { W[i] = X[i]; Inv[i] = 0.0f; }
        for (int i = 0; i < 8; ++i) Inv[i * 8 + i] = 1.0f;
        for (int col = 0; col < 8; ++col) {
            int piv = col; float pv = fabsf(W[col * 8 + col]);
            for (int r = col + 1; r < 8; ++r) {
                const float v = fabsf(W[r * 8 + col]);
                if (v > pv) { pv = v; piv = r; }
            }
            if (piv != col) {
                for (int c = 0; c < 8; ++c) {
                    float tw = W[col*8+c];  W[col*8+c]  = W[piv*8+c];  W[piv*8+c]  = tw;
                    float ti = Inv[col*8+c]; Inv[col*8+c] = Inv[piv*8+c]; Inv[piv*8+c] = ti;
                }
            }
            float d = W[col * 8 + col];
            d = (fabsf(d) > 1e-30f) ? 1.0f / d : 0.0f;
            for (int c = 0; c < 8; ++c) { W[col*8+c] *= d; Inv[col*8+c] *= d; }
            for (int r = 0; r < 8; ++r) {
                if (r == col) continue;
                const float f = W[r * 8 + col];
                for (int c = 0; c < 8; ++c) { W[r*8+c] -= f * W[col*8+c]; Inv[r*8+c] -= f * Inv[col*8+c]; }
            }
        }
        for (int r = 0; r < 8; ++r)
            for (int c = 0; c < 8; ++c)
                X[r * 8 + c] = 0.5f * (X[r * 8 + c] + Inv[c * 8 + r]);
    }
    if (transpose_out) {
        for (int r = 0; r < 8; ++r)
            for (int c = 0; c < 8; ++c) Qout[r * 8 + c] = X[c * 8 + r];
    } else {
        for (int i = 0; i < 64; ++i) Qout[i] = X[i];
    }
}

// ---------------------------------------------------------------------------
// Sequential Gauss-Seidel F sweep (single workgroup; G register-resident).
// ---------------------------------------------------------------------------
__global__ __launch_bounds__(1024) void f_sweep_kernel(const float* __restrict__ Gg,
                                                       const float* __restrict__ SPT,
                                                       const float* __restrict__ dsp,
                                                       const float* __restrict__ UQ,
                                                       float* __restrict__ F,
                                                       int n, int sweeps) {
    const int tid = threadIdx.x;
    float G0[KC], G1[KC];
    __shared__ float delta[KC];
#pragma unroll
    for (int k = 0; k < KC; ++k) {
        G0[k] = Gg[(size_t)tid * KC + k];
        G1[k] = Gg[(size_t)(tid + 1024) * KC + k];
    }
    for (int sw = 0; sw < sweeps; ++sw) {
        for (int i = 0; i < n; ++i) {
            if (tid == (i & 1023)) {
                const float d = dsp[i];
                float e[KC]; float emin = 3.4e38f;
#pragma unroll
                for (int k = 0; k < KC; ++k) {
                    const float gk = (i < 1024) ? G0[k] : G1[k];
                    const float fk = F[(size_t)i * KC + k];
                    e[k] = d + 2.0f * (gk - d * fk - ALPHA * UQ[(size_t)i * KC + k]);
                    emin = fminf(emin, e[k]);
                }
#pragma unroll
                for (int k = 0; k < KC; ++k) {
                    const float fk = F[(size_t)i * KC + k];
                    const float nv = ((e[k] == emin) || (e[k] < 0.0f)) ? 1.0f : 0.0f;
                    delta[k] = nv - fk;
                    F[(size_t)i * KC + k] = nv;
                }
            }
            __syncthreads();
            float dl[KC];
#pragma unroll
            for (int k = 0; k < KC; ++k) dl[k] = delta[k];
            const float* sprow = SPT + (size_t)i * n;   // column i of S'
            const float sp0 = sprow[tid];
            const float sp1 = sprow[tid + 1024];
#pragma unroll
            for (int k = 0; k < KC; ++k) { G0[k] += sp0 * dl[k]; G1[k] += sp1 * dl[k]; }
            __syncthreads();
        }
    }
}

// ---------------------------------------------------------------------------
// Host orchestration
// ---------------------------------------------------------------------------
static inline void skinny_gemm(const float* A, const float* B, float* C, float* Cpart,
                               hipStream_t stream) {
    wmma_skinny_gemm<<<dim3(N / 16, KSPLIT), 32, 0, stream>>>(A, B, Cpart, N);
    skinny_reduce_kernel<<<(N * KC + 255) / 256, 256, 0, stream>>>(Cpart, C, N * KC);
}

extern "C" void kernel_launch(void* const* d_in, const int* in_sizes, int n_in,
                              void* d_out, int out_size, void* d_ws, size_t ws_size,
                              hipStream_t stream) {
    const float* A  = (const float*)d_in[0];
    const float* U0 = (const float*)d_in[1];
    const float* F0 = (const float*)d_in[2];
    const float* Q0 = (const float*)d_in[3];
    float* F = (float*)d_out;

    const size_t NN = (size_t)N * N;
    float* ws    = (float*)d_ws;
    float* bufM  = ws;            // AA -> K -> M -> S'
    float* bufX0 = ws + NN;
    float* bufX1 = ws + 2 * NN;
    float* bufT  = ws + 3 * NN;
    float* p = ws + 4 * NN;
    float* rs   = p; p += N;
    float* cs   = p; p += N;
    float* rm   = p; p += N;
    float* cm   = p; p += N;
    float* scal = p; p += 8;
    float* dsp  = p; p += N;
    float* Ua   = p; p += N * KC;
    float* Ub   = p; p += N * KC;
    float* AU   = p; p += N * KC;
    float* UQ   = p; p += N * KC;
    float* Gg   = p; p += N * KC;
    float* FQp  = p; p += N * KC;
    float* FQm  = p; p += N * KC;
    float* Qw   = p; p += 64;
    float* Qp   = p; p += 64;
    float* Qm   = p; p += 64;
    float* UtU  = p; p += 64;
    float* UtF  = p; p += 64;
    float* Cpart = p; p += (size_t)KSPLIT * N * KC;

    (void)hipMemcpyAsync(F,  F0, (size_t)N * KC * sizeof(float), hipMemcpyDeviceToDevice, stream);
    (void)hipMemcpyAsync(Ua, U0, (size_t)N * KC * sizeof(float), hipMemcpyDeviceToDevice, stream);
    (void)hipMemcpyAsync(Qw, Q0, 64 * sizeof(float),             hipMemcpyDeviceToDevice, stream);

    const dim3 gGemm(N / BN, N / BM);
    const dim3 gEw(N / 16, N / 16), bEw(16, 16);
    const dim3 gTr(N / 32, N / 32), bTr(32, 8);

    // ---- kernel matrix: K = exp(-0.5(rs_i + cs_j - 2 A@A)) ----
    wmma_gemm_nn<<<gGemm, 256, 0, stream>>>(A, A, bufM, N);
    row_reduce_kernel<<<N, 256, 0, stream>>>(A, rs, N, 0, 1.0f);
    col_reduce_kernel<<<N / 256, 256, 0, stream>>>(A, cs, N, 0, 1.0f);
    kexp_kernel<<<gEw, bEw, 0, stream>>>(bufM, rs, cs, N);

    // ---- M = H K H + gamma I (in place) ----
    row_reduce_kernel<<<N, 256, 0, stream>>>(bufM, rm, N, 1, 1.0f / N);
    col_reduce_kernel<<<N / 256, 256, 0, stream>>>(bufM, cm, N, 1, 1.0f / N);
    vec_reduce_kernel<<<1, 256, 0, stream>>>(rm, &scal[0], N, 0, 1.0f / N);
    center_kernel<<<gEw, bEw, 0, stream>>>(bufM, rm, cm, &scal[0], N);

    // ---- M^-1 via Newton-Schulz ----
    row_reduce_kernel<<<N, 256, 0, stream>>>(bufM, rm, N, 2, 1.0f);
    vec_reduce_kernel<<<1, 256, 0, stream>>>(rm, &scal[2], N, 1, 1.0f);
    col_reduce_kernel<<<N / 256, 256, 0, stream>>>(bufM, cm, N, 2, 1.0f);
    vec_reduce_kernel<<<1, 256, 0, stream>>>(cm, &scal[1], N, 1, 1.0f);
    ns_scale_kernel<<<1, 32, 0, stream>>>(scal, &scal[3]);
    transpose_scale_kernel<<<gTr, bTr, 0, stream>>>(bufM, &scal[3], bufX0, N);

    float* X  = bufX0;
    float* Xo = bufX1;
    for (int it = 0; it < NS_ITERS; ++it) {
        wmma_gemm_nn<<<gGemm, 256, 0, stream>>>(bufM, X, bufT, N);
        y_kernel<<<gEw, bEw, 0, stream>>>(bufT, N);
        wmma_gemm_nn<<<gGemm, 256, 0, stream>>>(X, bufT, Xo, N);
        float* tmp = X; X = Xo; Xo = tmp;
    }

    // ---- S' ----
    sprime_kernel<<<gEw, bEw, 0, stream>>>(X, bufM, Xo, dsp, N);
    float* SP  = bufM;
    float* SPT = Xo;

    // ---- Q = proj(Q0) ----
    polar8_kernel<<<1, 32, 0, stream>>>(Qw, Qw, 0);

    for (int outer = 0; outer < N_OUTER; ++outer) {
        qpm_kernel<<<1, 64, 0, stream>>>(Qw, Qp, Qm);
        fq_kernel<<<N / 256, 256, 0, stream>>>(F, Qp, Qm, FQp, FQm, N);
        float* Uc = Ua; float* Un = Ub;
        for (int s = 0; s < N_USTEP; ++s) {
            skinny_gemm(A, Uc, AU, Cpart, stream);
            atb8_kernel<<<1, 64, 0, stream>>>(Uc, Uc, UtU, N);
            u_update_kernel<<<(N * KC + 255) / 256, 256, 0, stream>>>(Uc, AU, FQp, FQm, UtU, Un, N);
            float* tmp = Uc; Uc = Un; Un = tmp;
        }
        uq_kernel<<<(N * KC + 255) / 256, 256, 0, stream>>>(Uc, Qw, UQ, N);
        skinny_gemm(SP, F, Gg, Cpart, stream);
        f_sweep_kernel<<<1, 1024, 0, stream>>>(Gg, SPT, dsp, UQ, F, N, N_INNER);
        atb8_kernel<<<1, 64, 0, stream>>>(Uc, F, UtF, N);
        polar8_kernel<<<1, 32, 0, stream>>>(UtF, Qw, 1);
    }
}